// Spherical_26456998543634
// MI455X (gfx1250) — compile-verified
//
#include <hip/hip_runtime.h>

typedef __attribute__((ext_vector_type(2))) float v2f;
typedef __attribute__((ext_vector_type(8))) float v8f;

// out = |s| * x, routed through the CDNA5 matrix pipe:
// per 16x16 fp32 tile X:  D = sum_j (cols[4j:4j+4) of s*I16) x (rows[4j:4j+4) of X)
//                           = s*I16*X = s*X   (bit-exact: one RNE multiply per element)
// 4 chained V_WMMA_F32_16X16X4_F32 per tile; NT b32 loads/stores, fully coalesced.
__global__ void spherical_scale_wmma(const float* __restrict__ x,
                                     const float* __restrict__ s_ptr,
                                     float* __restrict__ out,
                                     long n_tiles, long n_total) {
  const float s = __builtin_fabsf(s_ptr[0]);   // uniform -> s_load
  const int lane = threadIdx.x & 31;           // wave32
  const int col  = lane & 15;
  const int hi   = lane >> 4;

  // A_j = columns [4j, 4j+4) of s*I16 in the 16x4 f32 A layout:
  //   lanes 0-15: row=col, K=kk ; lanes 16-31: row=col, K=kk+2
  v2f a[4];
#pragma unroll
  for (int j = 0; j < 4; ++j) {
    a[j].x = (col == (4 * j + 0 + 2 * hi)) ? s : 0.0f;
    a[j].y = (col == (4 * j + 1 + 2 * hi)) ? s : 0.0f;
  }

  const long wave   = (long)blockIdx.x * (blockDim.x >> 5) + (threadIdx.x >> 5);
  const long nwaves = (long)gridDim.x * (blockDim.x >> 5);

  const int ld_off = 32 * hi + col;   // B layout: row kk (+2 for upper lanes), N=col
  const int st_off = 128 * hi + col;  // D layout: row v (+8 for upper lanes), N=col

  for (long t = wave; t < n_tiles; t += nwaves) {
    const float* p = x + t * 256 + ld_off;
    v2f b[4];
#pragma unroll
    for (int j = 0; j < 4; ++j) {   // 8 coalesced NT b32 loads (streams >> 192MB L2)
      b[j].x = __builtin_nontemporal_load(p + 64 * j);
      b[j].y = __builtin_nontemporal_load(p + 64 * j + 16);
    }

    v8f c = {};
#pragma unroll
    for (int j = 0; j < 4; ++j) {   // chained accumulation: D = s*I16*X = s*X
      c = __builtin_amdgcn_wmma_f32_16x16x4_f32(
              /*neg_a=*/false, a[j], /*neg_b=*/false, b[j],
              /*c_mod=*/(short)0, c, /*reuse_a=*/false, /*reuse_b=*/false);
    }

    float* q = out + t * 256 + st_off;
#pragma unroll
    for (int v = 0; v < 8; ++v)     // 8 coalesced NT b32 stores
      __builtin_nontemporal_store(c[v], q + 16 * v);
  }

  // Generic tail (N % 256 != 0); N = 128M is tile-exact so this is empty here.
  const long tail     = n_tiles * 256;
  const long tid      = (long)blockIdx.x * blockDim.x + threadIdx.x;
  const long nthreads = (long)gridDim.x * blockDim.x;
  for (long i = tail + tid; i < n_total; i += nthreads)
    out[i] = s * x[i];
}

extern "C" void kernel_launch(void* const* d_in, const int* in_sizes, int n_in,
                              void* d_out, int out_size, void* d_ws, size_t ws_size,
                              hipStream_t stream) {
  const float* x = (const float*)d_in[0];
  const float* s = (const float*)d_in[1];
  float* out     = (float*)d_out;

  const long n     = (long)out_size;   // == in_sizes[0] (flat element count of x)
  const long tiles = n / 256;

  // Persistent grid: 256 threads = 8 wave32 waves per block; cap blocks so each
  // wave iterates several tiles (amortize setup, keep tens of thousands of
  // waves in flight for HBM latency hiding at 23.3 TB/s).
  long blocks_l = (tiles + 7) / 8;
  if (blocks_l < 1)    blocks_l = 1;
  if (blocks_l > 6144) blocks_l = 6144;   // 6144 blocks * 8 waves = 49152 waves

  spherical_scale_wmma<<<(int)blocks_l, 256, 0, stream>>>(x, s, out, tiles, n);
}